// SNN_7876970021477
// MI455X (gfx1250) — compile-verified
//
#include <hip/hip_runtime.h>
#include <hip/hip_bf16.h>

typedef float v2f __attribute__((ext_vector_type(2)));
typedef float v8f __attribute__((ext_vector_type(8)));

#define B_ 256
#define T_ 100
#define F_ 784
#define H_ 512
#define O_ 10
#define ALPHA_ 0.9048374180359595f
#define BETA_  0.8187307530779818f

// LDS row strides (padded for conflict-free ds_load_b64 A-fragments)
#define XS 788   // 788 % 64 = 20 -> 16 distinct banks across lanes
#define ZS 516   // 516 % 64 = 4  -> 16 distinct banks across lanes

// packed-weight regions in d_ws (float offsets)
#define PWF0_OFF 0
#define PWR0_OFF (F_ * H_)
#define PWF1_OFF (PWR0_OFF + H_ * H_)
#define PWR1_OFF (PWF1_OFF + H_ * H_)
#define PACK_FLOATS (PWR1_OFF + H_ * H_)

static __device__ __forceinline__ v8f wmma4(v2f a, v2f b, v8f c) {
  // V_WMMA_F32_16X16X4_F32 : D(16x16 f32) = A(16x4 f32) * B(4x16 f32) + C
  return __builtin_amdgcn_wmma_f32_16x16x4_f32(
      false, a, false, b, (short)0, c, false, false);
}

// K-sweep of one 16x16 output tile.
// arow: LDS A-row pointer already offset by ln*stride + 2*hi.
// PACKED: W is P[g][n][4] with P[(g*H + n)*4 + j] = Worig[4g + j][n]
//         -> one coalesced global_load_b64 per WMMA, constant-stride offsets.
// else:   W is the original row-major [K][H_] weight.
template <int K, bool PACKED>
static __device__ __forceinline__ void gemm_k(const float* __restrict__ arow,
                                              const float* __restrict__ W,
                                              int col, int hi, v8f& acc) {
  if (PACKED) {
    const float* bp = W + (size_t)col * 4 + 2 * hi;
#pragma unroll 8
    for (int g = 0; g < K / 4; ++g) {
      v2f a = *(const v2f*)(arow + 4 * g);
      v2f b = *(const v2f*)(bp + (size_t)g * (H_ * 4));
      acc = wmma4(a, b, acc);
    }
  } else {
#pragma unroll 4
    for (int k0 = 0; k0 < K; k0 += 4) {
      v2f a = *(const v2f*)(arow + k0);
      int kb = k0 + 2 * hi;
      v2f b;
      b.x = W[(size_t)kb * H_ + col];
      b.y = W[(size_t)(kb + 1) * H_ + col];
      acc = wmma4(a, b, acc);
    }
  }
}

// repack W[K][N] -> P[((k>>2)*N + n)*4 + (k&3)]
__global__ void pack_weights_kernel(const float* __restrict__ W,
                                    float* __restrict__ P, int K, int N) {
  int i = blockIdx.x * 256 + threadIdx.x;
  if (i < K * N) {
    int k = i / N, n = i - k * N;
    P[((size_t)(k >> 2) * N + n) * 4 + (k & 3)] = W[i];
  }
}

template <bool PACKED>
__global__ __launch_bounds__(256, 1) void snn_scan_kernel(
    const float* __restrict__ x,    // [B,T,F]
    const float* __restrict__ Wf0,  // [F,H]   (packed layout if PACKED)
    const float* __restrict__ Wr0,  // [H,H]
    const float* __restrict__ Wf1,  // [H,H]
    const float* __restrict__ Wr1,  // [H,H]
    const float* __restrict__ Wro,  // [H,O]   (always raw)
    float* __restrict__ out)        // [B,T,O]
{
  extern __shared__ float smem[];
  float* xb   = smem;                        // [2][16][XS]   x_t double buffer
  float* z0b  = xb  + 2 * 16 * XS;           // [2][16][ZS]   Z0 double buffer
  float* z1b  = z0b + 2 * 16 * ZS;           // [2][16][ZS]   Z1 double buffer
  float* wroS = z1b + 2 * 16 * ZS;           // [512][16]     Wro padded to 16 cols
  float* red  = wroS + H_ * 16;              // [8][256]      readout partials

  const int tid  = threadIdx.x;
  const int wave = tid >> 5;
  const int lane = tid & 31;
  const int hi   = lane >> 4;   // half-wave select
  const int ln   = lane & 15;   // row (A) / col (B,C,D) index
  const int b0   = blockIdx.x * 16;

  // ---- one-time staging: Wro (zero-padded cols), zero parity-0 Z, load x(t=0)
  for (int i = tid; i < H_ * 16; i += 256) {
    int k = i >> 4, n = i & 15;
    wroS[i] = (n < O_) ? Wro[k * O_ + n] : 0.0f;
  }
  for (int i = tid; i < 16 * ZS; i += 256) { z0b[i] = 0.0f; z1b[i] = 0.0f; }
  for (int i = tid; i < 16 * F_; i += 256) {
    int m = i / F_, k = i - m * F_;
    xb[m * XS + k] = x[(size_t)(b0 + m) * T_ * F_ + k];
  }
  __syncthreads();

  v8f I0[4], V0[4], I1[4], V1[4];
  v8f Ir = {0.f,0.f,0.f,0.f,0.f,0.f,0.f,0.f};
  v8f Vr = {0.f,0.f,0.f,0.f,0.f,0.f,0.f,0.f};
#pragma unroll
  for (int c = 0; c < 4; ++c) { I0[c] = Ir; V0[c] = Ir; I1[c] = Ir; V1[c] = Ir; }

  for (int t = 0; t < T_; ++t) {
    const int p = t & 1;
    float* xcur = xb  + p       * (16 * XS);
    float* xnxt = xb  + (1 - p) * (16 * XS);
    float* z0p  = z0b + p       * (16 * ZS);
    float* z0n  = z0b + (1 - p) * (16 * ZS);
    float* z1p  = z1b + p       * (16 * ZS);
    float* z1n  = z1b + (1 - p) * (16 * ZS);

    // ---- async prefetch x(t+1) straight into LDS (ASYNCcnt-tracked);
    //      drained by s_wait_asynccnt before the end-of-step barrier.
    if (t + 1 < T_) {
      unsigned long long xbase =
          (unsigned long long)(uintptr_t)(x + (size_t)(t + 1) * F_);
      for (int i = tid; i < 16 * F_; i += 256) {
        int m = i / F_, k = i - m * F_;
        unsigned ldsoff =
            (unsigned)((size_t)((xnxt - smem) + m * XS + k) * sizeof(float));
        unsigned goff = (unsigned)(((size_t)(b0 + m) * T_ * F_ + k) * sizeof(float));
        asm volatile("global_load_async_to_lds_b32 %0, %1, %2"
                     :
                     : "v"(ldsoff), "v"(goff), "s"(xbase)
                     : "memory");
      }
    }

    // ================= layer 0 : I0 = a*I0 + x@Wf0 + Z0@Wr0 =================
#pragma unroll
    for (int c = 0; c < 4; ++c) {
      const int col = wave * 64 + c * 16 + ln;
      v8f acc = {0.f,0.f,0.f,0.f,0.f,0.f,0.f,0.f};
      gemm_k<F_, PACKED>(xcur + ln * XS + 2 * hi, Wf0, col, hi, acc);
      gemm_k<H_, PACKED>(z0p  + ln * ZS + 2 * hi, Wr0, col, hi, acc);
#pragma unroll
      for (int v = 0; v < 8; ++v) {
        int m = v + 8 * hi;
        float zp = z0p[m * ZS + col];
        float i0 = ALPHA_ * I0[c][v] + acc[v];
        float v0 = (BETA_ * V0[c][v] + i0) * (1.0f - zp);
        I0[c][v] = i0; V0[c][v] = v0;
        z0n[m * ZS + col] = (v0 - 1.0f > 0.0f) ? 1.0f : 0.0f;
      }
    }
    __syncthreads();   // all Z0(new) visible

    // ================= layer 1 : I1 = a*I1 + Z0n@Wf1 + Z1@Wr1 ===============
#pragma unroll
    for (int c = 0; c < 4; ++c) {
      const int col = wave * 64 + c * 16 + ln;
      v8f acc = {0.f,0.f,0.f,0.f,0.f,0.f,0.f,0.f};
      gemm_k<H_, PACKED>(z0n + ln * ZS + 2 * hi, Wf1, col, hi, acc);
      gemm_k<H_, PACKED>(z1p + ln * ZS + 2 * hi, Wr1, col, hi, acc);
#pragma unroll
      for (int v = 0; v < 8; ++v) {
        int m = v + 8 * hi;
        float zp = z1p[m * ZS + col];
        float i1 = ALPHA_ * I1[c][v] + acc[v];
        float v1 = (BETA_ * V1[c][v] + i1) * (1.0f - zp);
        I1[c][v] = i1; V1[c][v] = v1;
        z1n[m * ZS + col] = (v1 - 1.0f > 0.0f) ? 1.0f : 0.0f;
      }
    }
    __syncthreads();   // all Z1(new) visible

    // ================= readout : Ir = a*Ir + Z1n@Wro ; Vr = b*Vr + Ir =======
    {
      v8f acc = {0.f,0.f,0.f,0.f,0.f,0.f,0.f,0.f};
      const float* zrow = z1n + ln * ZS + wave * 64 + 2 * hi;  // K-slice per wave
#pragma unroll
      for (int k0 = 0; k0 < 64; k0 += 4) {
        v2f a = *(const v2f*)(zrow + k0);
        int kb = wave * 64 + k0 + 2 * hi;
        v2f b;
        b.x = wroS[kb * 16 + ln];
        b.y = wroS[(kb + 1) * 16 + ln];
        acc = wmma4(a, b, acc);
      }
#pragma unroll
      for (int v = 0; v < 8; ++v) red[wave * 256 + v * 32 + lane] = acc[v];
    }
    __syncthreads();
#pragma unroll
    for (int v = 0; v < 8; ++v) {
      float s = 0.0f;
#pragma unroll
      for (int w2 = 0; w2 < 8; ++w2) s += red[w2 * 256 + v * 32 + lane];
      float ir = ALPHA_ * Ir[v] + s;
      float vr = BETA_ * Vr[v] + ir;
      Ir[v] = ir; Vr[v] = vr;
      if (wave == 0 && ln < O_) {
        int m = v + 8 * hi;
        out[((size_t)(b0 + m) * T_ + t) * O_ + ln] = vr;
      }
    }
    // drain async x prefetch, then step boundary
    asm volatile("s_wait_asynccnt 0" ::: "memory");
    __syncthreads();
  }
}

extern "C" void kernel_launch(void* const* d_in, const int* in_sizes, int n_in,
                              void* d_out, int out_size, void* d_ws, size_t ws_size,
                              hipStream_t stream) {
  (void)in_sizes; (void)n_in; (void)out_size;
  const float* x   = (const float*)d_in[0];
  const float* Wf0 = (const float*)d_in[1];
  const float* Wr0 = (const float*)d_in[2];
  const float* Wf1 = (const float*)d_in[3];
  const float* Wr1 = (const float*)d_in[4];
  const float* Wro = (const float*)d_in[5];
  float* out = (float*)d_out;

  const size_t smem =
      (size_t)(2 * 16 * XS + 2 * 16 * ZS + 2 * 16 * ZS + H_ * 16 + 8 * 256) *
      sizeof(float);  // ~273,920 bytes < 320 KB LDS per WGP

  const bool use_packed = ws_size >= (size_t)PACK_FLOATS * sizeof(float);

  if (use_packed) {
    float* P = (float*)d_ws;
    pack_weights_kernel<<<dim3((F_ * H_ + 255) / 256), dim3(256), 0, stream>>>(
        Wf0, P + PWF0_OFF, F_, H_);
    pack_weights_kernel<<<dim3((H_ * H_ + 255) / 256), dim3(256), 0, stream>>>(
        Wr0, P + PWR0_OFF, H_, H_);
    pack_weights_kernel<<<dim3((H_ * H_ + 255) / 256), dim3(256), 0, stream>>>(
        Wf1, P + PWF1_OFF, H_, H_);
    pack_weights_kernel<<<dim3((H_ * H_ + 255) / 256), dim3(256), 0, stream>>>(
        Wr1, P + PWR1_OFF, H_, H_);
    snn_scan_kernel<true><<<dim3(B_ / 16), dim3(256), smem, stream>>>(
        x, P + PWF0_OFF, P + PWR0_OFF, P + PWF1_OFF, P + PWR1_OFF, Wro, out);
  } else {
    snn_scan_kernel<false><<<dim3(B_ / 16), dim3(256), smem, stream>>>(
        x, Wf0, Wr0, Wf1, Wr1, Wro, out);
  }
}